// DiscreteContinuousEncoder_48069273977525
// MI455X (gfx1250) — compile-verified
//
#include <hip/hip_runtime.h>
#include <hip/hip_bf16.h>

// ---------------------------------------------------------------------------
// DISCO spherical conv encoder, fused two-stage GEMM on CDNA5 WMMA (fp32).
//   stage A: z[c*9+k][w]  = sum_n psi[k,h,n] * x[c, hi[n], (wi[n]+2w)%1440]
//   stage B: y[o,h,w]     = sum_ck weight[o][ck] * z[ck][w]
// One workgroup per (h, 48-wide w tile). z staged in LDS.
// The gather offset pattern is c-invariant, so it is materialized once per
// workgroup as an int32 LDS table; the stage-A inner step is then
// 2x ds_load (imm offset) + 2x v_add + 2x global_load + 1x v_wmma.
// fp32 matrix path: V_WMMA_F32_16X16X4_F32 (inputs fp32; kernel is
// gather/HBM-bound at ~10GB traffic vs 131 GFLOP, so no precision drop).
// ---------------------------------------------------------------------------

typedef __attribute__((ext_vector_type(2))) float v2f;
typedef __attribute__((ext_vector_type(8))) float v8f;

#define CIN   73
#define HIN   721
#define WIN   1440
#define COUT  256
#define KW    9
#define HOUT  361
#define NQ    128
#define WOUT  720
#define WT    48            // w-tile per workgroup (3 WMMA tiles of 16)
#define CK    (CIN * KW)    // 657
#define CKP   660           // padded to multiple of 4 for clean K-steps

#define Z_BYTES   (CKP * WT * 4)        // 126720
#define OFF_BYTES (NQ * WT * 4)         // 24576  int32 gather offsets
#define PSI_BYTES (16 * NQ * 4)         // 8192   (rows 9..15 zero-padded)
#define SMEM_BYTES (Z_BYTES + OFF_BYTES + PSI_BYTES + NQ * 4 + NQ * 4) // 160512

__global__ __launch_bounds__(256)
void disco_encoder_kernel(const float* __restrict__ x,
                          const float* __restrict__ weight,
                          const float* __restrict__ psi,
                          const int*   __restrict__ idx_hi,
                          const int*   __restrict__ idx_wi,
                          float*       __restrict__ y)
{
    extern __shared__ char smem[];
    float* zs   = (float*)smem;                                   // [CKP][WT]
    int*   offt = (int*)  (smem + Z_BYTES);                       // [NQ][WT]
    float* psis = (float*)(smem + Z_BYTES + OFF_BYTES);           // [16][NQ]
    int*   hio  = (int*)  (smem + Z_BYTES + OFF_BYTES + PSI_BYTES);
    int*   wio  = hio + NQ;

    const int h    = blockIdx.y;
    const int w0   = blockIdx.x * WT;
    const int tid  = threadIdx.x;
    const int lane = tid & 31;
    const int wave = tid >> 5;                // 0..7
    const int mrow = lane & 15;               // M row within 16x16 tile
    const int half = lane >> 4;               // 0 / 1
    const int khalf = half << 1;              // K offset 0 / 2 (A & B frag layout)
    const int mhi  = half * 8;                // C/D: M = r + 8*half

    // ---- phase 1: metadata into LDS ------------------------------------
    for (int i = tid; i < 16 * NQ; i += 256) {
        int k = i >> 7, n = i & 127;
        psis[i] = (k < KW) ? psi[(k * HOUT + h) * NQ + n] : 0.0f;
    }
    for (int i = tid; i < NQ; i += 256) {
        hio[i] = idx_hi[h * NQ + i] * WIN;
        wio[i] = idx_wi[h * NQ + i];
    }
    for (int i = tid; i < (CKP - CK) * WT; i += 256)          // zero K-pad rows
        zs[CK * WT + i] = 0.0f;
    __syncthreads();

    // ---- phase 2: c-invariant gather-offset table -----------------------
    // offt[n][wc] = hi[n]*WIN + (wi[n] + 2*(w0+wc)) % WIN   (built once,
    // reused by all 73 channels -> removes wrap VALU from the hot loop)
    for (int i = tid; i < NQ * WT; i += 256) {
        int n  = i / WT;
        int wc = i - n * WT;
        int wr = wio[n] + 2 * (w0 + wc);
        if (wr >= WIN) wr -= WIN;
        offt[i] = hio[n] + wr;
    }

    // Preload all 32 A-fragments (psi shared by every c and every wtile).
    v2f afrag[32];
#pragma unroll
    for (int s = 0; s < 32; ++s) {
        int n0 = 4 * s + khalf;
        afrag[s].x = psis[mrow * NQ + n0];
        afrag[s].y = psis[mrow * NQ + n0 + 1];
    }
    __syncthreads();

    // ---- Stage A: psi(16x128, padded) x gathered(128x16) -> z tiles ----
    for (int job = wave; job < CIN * 3; job += 8) {           // 219 jobs / 8 waves
        const int c  = job / 3;
        const int wt = job - 3 * c;
        const int wcol = wt * 16 + mrow;                      // lane's N column
        const int* offc = offt + wcol;                        // base + imm offsets
        const int cbase = c * (HIN * WIN);
        v8f acc = {};
#pragma unroll
        for (int s = 0; s < 32; ++s) {
            const int n0 = 4 * s + khalf;
            v2f b;
            b.x = x[cbase + offc[n0 * WT]];
            b.y = x[cbase + offc[(n0 + 1) * WT]];
            acc = __builtin_amdgcn_wmma_f32_16x16x4_f32(
                      false, afrag[s], false, b, (short)0, acc, false, false);
        }
        // D rows M = k (only 0..8 are real); store to z[(c*9+k)][w]
#pragma unroll
        for (int r = 0; r < 8; ++r) {
            int k = r + mhi;
            if (k < KW) zs[(c * KW + k) * WT + wcol] = acc[r];
        }
    }
    __syncthreads();

    // ---- Stage B: weight(256x657) x z(657x48) -> y tile ----------------
    const int ot0 = wave * 2;                 // 2 o-tiles per wave
    v8f accB[2][3] = {};

    for (int s = 0; s < 164; ++s) {           // ck 0..655, unguarded
        const int ck0 = 4 * s + khalf;
        v2f bfrag[3];
#pragma unroll
        for (int wt = 0; wt < 3; ++wt) {
            int wc = wt * 16 + mrow;
            bfrag[wt].x = zs[ck0 * WT + wc];
            bfrag[wt].y = zs[(ck0 + 1) * WT + wc];
        }
#pragma unroll
        for (int t = 0; t < 2; ++t) {
            const int o = (ot0 + t) * 16 + mrow;
            v2f a;
            a.x = weight[o * CK + ck0];
            a.y = weight[o * CK + ck0 + 1];
#pragma unroll
            for (int wt = 0; wt < 3; ++wt)
                accB[t][wt] = __builtin_amdgcn_wmma_f32_16x16x4_f32(
                    false, a, false, bfrag[wt], (short)0, accB[t][wt], false, false);
        }
    }
    {   // peeled final K-step: ck 656..659 (z rows >=657 are zero, guard weight)
        const int ck0 = 656 + khalf;
        v2f bfrag[3];
#pragma unroll
        for (int wt = 0; wt < 3; ++wt) {
            int wc = wt * 16 + mrow;
            bfrag[wt].x = zs[ck0 * WT + wc];
            bfrag[wt].y = zs[(ck0 + 1) * WT + wc];
        }
#pragma unroll
        for (int t = 0; t < 2; ++t) {
            const int o = (ot0 + t) * 16 + mrow;
            v2f a;
            a.x = (ck0     < CK) ? weight[o * CK + ck0]     : 0.0f;
            a.y = (ck0 + 1 < CK) ? weight[o * CK + ck0 + 1] : 0.0f;
#pragma unroll
            for (int wt = 0; wt < 3; ++wt)
                accB[t][wt] = __builtin_amdgcn_wmma_f32_16x16x4_f32(
                    false, a, false, bfrag[wt], (short)0, accB[t][wt], false, false);
        }
    }

    // ---- epilogue: y[o, h, w] ------------------------------------------
#pragma unroll
    for (int t = 0; t < 2; ++t) {
#pragma unroll
        for (int wt = 0; wt < 3; ++wt) {
            const int wcol = w0 + wt * 16 + mrow;
#pragma unroll
            for (int r = 0; r < 8; ++r) {
                const int o = (ot0 + t) * 16 + r + mhi;
                y[((long)o * HOUT + h) * WOUT + wcol] = accB[t][wt][r];
            }
        }
    }
}

extern "C" void kernel_launch(void* const* d_in, const int* in_sizes, int n_in,
                              void* d_out, int out_size, void* d_ws, size_t ws_size,
                              hipStream_t stream) {
    const float* x      = (const float*)d_in[0];   // (1,73,721,1440)
    const float* weight = (const float*)d_in[1];   // (256,73,9)
    const float* psi    = (const float*)d_in[2];   // (9,361,128)
    const int*   idx_hi = (const int*)  d_in[3];   // (361,128)
    const int*   idx_wi = (const int*)  d_in[4];   // (361,128)
    float*       y      = (float*)d_out;           // (1,256,361,720)

    // 160512 B dynamic LDS per workgroup -> 2 workgroups per 320KB WGP
    hipFuncSetAttribute((const void*)disco_encoder_kernel,
                        hipFuncAttributeMaxDynamicSharedMemorySize, SMEM_BYTES);

    dim3 grid(WOUT / WT, HOUT);   // (15, 361)
    disco_encoder_kernel<<<grid, 256, SMEM_BYTES, stream>>>(
        x, weight, psi, idx_hi, idx_wi, y);
}